// VectorQuantizer2_44495861187029
// MI455X (gfx1250) — compile-verified
//
#include <hip/hip_runtime.h>
#include <hip/hip_bf16.h>

typedef __attribute__((ext_vector_type(2))) float v2f;
typedef __attribute__((ext_vector_type(4))) float v4f;
typedef __attribute__((ext_vector_type(8))) float v8f;
typedef __attribute__((ext_vector_type(4))) int   v4i;

#define VQ_D        256
#define VQ_HW       1024
#define VQ_N        16384          // 16 * 32 * 32
#define VQ_K        8192
#define VQ_NT       (VQ_K / 16)    // 512 codebook tiles
#define ROWS_PER_WG 128
#define LDSB_STRIDE 260            // 256 + 4 pad: kills stride-256 bank conflicts

#if __has_builtin(__builtin_amdgcn_global_load_async_to_lds_b128)
#define VQ_HAS_ASYNC 1
#endif

// d_ws layout (floats): [0] = loss accumulator, [64 .. 64+8192) = codebook norms

// ---------------------------------------------------------------------------
// Kernel 1: ||e_k||^2 for all codebook rows; also zero the loss accumulator.
// ---------------------------------------------------------------------------
__global__ __launch_bounds__(256) void vq_norms(const float* __restrict__ cb,
                                                float* __restrict__ ws) {
    if (blockIdx.x == 0 && threadIdx.x == 0) ws[0] = 0.0f;
    const int w    = threadIdx.x >> 5;
    const int lane = threadIdx.x & 31;
    const int k    = blockIdx.x * 8 + w;          // grid = K/8 blocks
    const float* row = cb + (size_t)k * VQ_D;
    float s = 0.0f;
#pragma unroll
    for (int i = 0; i < VQ_D / 32; ++i) {
        float v = row[lane + i * 32];
        s += v * v;
    }
#pragma unroll
    for (int off = 16; off >= 1; off >>= 1) s += __shfl_down(s, off, 32);
    if (lane == 0) ws[64 + k] = s;
}

// ---------------------------------------------------------------------------
// Async staging helpers (CDNA5 ASYNCcnt path)
// ---------------------------------------------------------------------------
__device__ __forceinline__ void vq_wait_async0() {
#if __has_builtin(__builtin_amdgcn_s_wait_asynccnt)
    __builtin_amdgcn_s_wait_asynccnt(0);
#else
    asm volatile("s_wait_asynccnt 0x0" ::: "memory");
#endif
}

#ifdef VQ_HAS_ASYNC
// Builtin signature (from clang diagnostic): param0 = int4 addrspace(1)*,
// param1 = int4 addrspace(3)*, then imm offset, imm cpol.
typedef __attribute__((address_space(1))) v4i vq_gi4;
typedef __attribute__((address_space(3))) v4i vq_li4;
// Each thread copies 16 contiguous floats (4 x b128) global -> LDS, no VGPR
// round trip; tracked by ASYNCcnt.
__device__ __forceinline__ void vq_stage_async(const float* src, float* dst) {
#pragma unroll
    for (int i = 0; i < 16; i += 4) {
        __builtin_amdgcn_global_load_async_to_lds_b128(
            (vq_gi4*)(src + i), (vq_li4*)(dst + i), 0, 0);
    }
}
#endif

// ---------------------------------------------------------------------------
// Kernel 2: distances via f32 WMMA, argmin, codebook gather, fused loss.
// 8 waves/WG, each wave owns a 16-row A tile in registers as WMMA A-frags.
// B tiles double-buffered in LDS, staged asynchronously ahead of compute.
// ---------------------------------------------------------------------------
__global__ __launch_bounds__(256) void vq_main(const float* __restrict__ z,
                                               const float* __restrict__ cb,
                                               const float* __restrict__ norms,
                                               float* __restrict__ loss_accum,
                                               float* __restrict__ out) {
    __shared__ float lds_b[2][16 * LDSB_STRIDE];
    __shared__ int   lds_idx[ROWS_PER_WG];
    __shared__ float lds_red[8];

    const int tid  = threadIdx.x;
    const int w    = tid >> 5;
    const int lane = tid & 31;
    const int hi   = lane >> 4;        // K-pair select (A/B); M/N half (C)
    const int lo   = lane & 15;
    const int row_base = blockIdx.x * ROWS_PER_WG;

    // ---- A fragments: 16 rows x 256 f32, WMMA 16x16x4 layout, in registers --
    const int m_a  = row_base + w * 16 + lo;
    const int b_a  = m_a >> 10;
    const int hw_a = m_a & 1023;
    const float* zrow = z + (size_t)b_a * (VQ_D * VQ_HW) + hw_a;
    v2f afrag[64];
#pragma unroll
    for (int kb = 0; kb < 64; ++kb) {
        const int d0 = kb * 4 + hi * 2;
        afrag[kb].x = zrow[(size_t)(d0 + 0) * VQ_HW];
        afrag[kb].y = zrow[(size_t)(d0 + 1) * VQ_HW];
    }

    // ---- per-thread staging coordinates (16 floats each, coalesced) ----
    const int brow = tid >> 4;
    const int bcol = (tid & 15) * 16;
    const float* cbsrc = cb + (size_t)brow * VQ_D + bcol;   // + ct*16*VQ_D
    float* ldst[2] = { &lds_b[0][brow * LDSB_STRIDE + bcol],
                       &lds_b[1][brow * LDSB_STRIDE + bcol] };

    float best_v[8];
    int   best_i[8];
#pragma unroll
    for (int r = 0; r < 8; ++r) { best_v[r] = 3.4e38f; best_i[r] = 0; }

#ifdef VQ_HAS_ASYNC
    vq_stage_async(cbsrc, ldst[0]);                  // prologue: tile 0
#else
    v4f pre[4];
#pragma unroll
    for (int i = 0; i < 4; ++i)
        pre[i] = *reinterpret_cast<const v4f*>(cbsrc + i * 4);
#endif

    for (int ct = 0; ct < VQ_NT; ++ct) {
        const int cur = ct & 1;
#ifdef VQ_HAS_ASYNC
        vq_wait_async0();                 // my slice of tile ct landed in LDS
        __syncthreads();                  // whole WG's tile ct visible; buf[1-cur] free
        if (ct + 1 < VQ_NT)               // overlap tile ct+1 with compute of ct
            vq_stage_async(cbsrc + (size_t)(ct + 1) * (16 * VQ_D), ldst[1 - cur]);
#else
        __syncthreads();                  // buf[cur] no longer being read
#pragma unroll
        for (int i = 0; i < 4; ++i)
            *reinterpret_cast<v4f*>(ldst[cur] + i * 4) = pre[i];
        if (ct + 1 < VQ_NT) {
            const float* nsrc = cbsrc + (size_t)(ct + 1) * (16 * VQ_D);
#pragma unroll
            for (int i = 0; i < 4; ++i)
                pre[i] = *reinterpret_cast<const v4f*>(nsrc + i * 4);
        }
        __syncthreads();                  // stores visible to all waves
#endif

        // 16x16 dot tile: 64 x v_wmma_f32_16x16x4_f32, dual accumulators to
        // break the C->D dependency chain between consecutive WMMAs.
        const float en = norms[ct * 16 + lo];       // L2-resident
        v8f acc0 = {}, acc1 = {};
        const float* bptr = &lds_b[cur][lo * LDSB_STRIDE + hi * 2];
#pragma unroll
        for (int kb = 0; kb < 64; kb += 2) {
            v2f bf0 = *reinterpret_cast<const v2f*>(bptr + kb * 4);
            v2f bf1 = *reinterpret_cast<const v2f*>(bptr + kb * 4 + 4);
            acc0 = __builtin_amdgcn_wmma_f32_16x16x4_f32(
                false, afrag[kb + 0], false, bf0, (short)0, acc0, false, false);
            acc1 = __builtin_amdgcn_wmma_f32_16x16x4_f32(
                false, afrag[kb + 1], false, bf1, (short)0, acc1, false, false);
        }

        // dist' = ||e||^2 - 2 z.e   (||z||^2 constant per row: drop it)
        const int n = ct * 16 + lo;
#pragma unroll
        for (int r = 0; r < 8; ++r) {
            const float dist = en - 2.0f * (acc0[r] + acc1[r]);
            if (dist < best_v[r]) { best_v[r] = dist; best_i[r] = n; }
        }
    }

    // ---- argmin reduction across the 16 lanes sharing each output row ----
    // C/D layout: VGPR r, lane l -> M = r + 8*(l>>4), N = l&15.
#pragma unroll
    for (int r = 0; r < 8; ++r) {
        float bv = best_v[r];
        int   bi = best_i[r];
#pragma unroll
        for (int off = 8; off >= 1; off >>= 1) {
            float ov = __shfl_xor(bv, off, 32);
            int   oi = __shfl_xor(bi, off, 32);
            if (ov < bv || (ov == bv && oi < bi)) { bv = ov; bi = oi; }
        }
        if (lo == 0) lds_idx[w * 16 + hi * 8 + r] = bi;
    }
    __syncthreads();

    // ---- epilogue: min_idx, z_q gather (straight-through == codebook row),
    //      fused loss partial sum ----
    int* midx = reinterpret_cast<int*>(out + (size_t)VQ_N * VQ_D + 1);
    if (tid < ROWS_PER_WG) midx[row_base + tid] = lds_idx[tid];

    const int rloc = tid & 127;
    const int dh   = tid >> 7;
    const int mrow = row_base + rloc;
    const int ob   = mrow >> 10;
    const int ohw  = mrow & 1023;
    const size_t obase = (size_t)ob * (VQ_D * VQ_HW) + ohw;
    const float* crow = cb + (size_t)lds_idx[rloc] * VQ_D;

    float lsum = 0.0f;
    for (int d0 = 0; d0 < VQ_D; d0 += 2) {
        const int d = d0 + dh;
        const float  v    = crow[d];
        const size_t addr = obase + (size_t)d * VQ_HW;   // hw-contiguous stores
        const float  diff = v - z[addr];
        out[addr] = v;
        lsum += diff * diff;
    }

#pragma unroll
    for (int off = 16; off >= 1; off >>= 1) lsum += __shfl_down(lsum, off, 32);
    if (lane == 0) lds_red[w] = lsum;
    __syncthreads();
    if (tid == 0) {
        float t = 0.0f;
#pragma unroll
        for (int i = 0; i < 8; ++i) t += lds_red[i];
        atomicAdd(loss_accum, t);
    }
}

// ---------------------------------------------------------------------------
// Kernel 3: loss = (1 + 0.25) * sum / (16*32*32*256)
// ---------------------------------------------------------------------------
__global__ void vq_finalize(const float* __restrict__ ws, float* __restrict__ out) {
    out[(size_t)VQ_N * VQ_D] = ws[0] * (1.25f / 4194304.0f);
}

extern "C" void kernel_launch(void* const* d_in, const int* in_sizes, int n_in,
                              void* d_out, int out_size, void* d_ws, size_t ws_size,
                              hipStream_t stream) {
    const float* z  = (const float*)d_in[0];   // [16,256,32,32]
    const float* cb = (const float*)d_in[1];   // [8192,256]
    float* out = (float*)d_out;
    float* ws  = (float*)d_ws;

    hipLaunchKernelGGL(vq_norms, dim3(VQ_K / 8), dim3(256), 0, stream, cb, ws);
    hipLaunchKernelGGL(vq_main, dim3(VQ_N / ROWS_PER_WG), dim3(256), 0, stream,
                       z, cb, ws + 64, ws, out);
    hipLaunchKernelGGL(vq_finalize, dim3(1), dim3(1), 0, stream, ws, out);
}